// Feat3dNet_27161373180518
// MI455X (gfx1250) — compile-verified
//
#include <hip/hip_runtime.h>
#include <math.h>

// ---------------- problem constants ----------------
#define BB   8
#define NN   16384
#define NPT  512
#define NSAMP 64
#define RAD_INV 0.5f
#define EPSV 1e-8f

// output layout (floats): new_xyz | feats | attention
#define OFF_NEWXYZ 0
#define OFF_FEATS  (BB*NPT*3)                 // 12288
#define OFF_ATT    (BB*NPT*3 + BB*NPT*64)     // 274432

typedef float v2f __attribute__((ext_vector_type(2)));
typedef float v8f __attribute__((ext_vector_type(8)));

// ---------------- WMMA f32 16x16x4 helpers ----------------
static __device__ __forceinline__ v8f wmma4(v8f c, v2f a, v2f b) {
  return __builtin_amdgcn_wmma_f32_16x16x4_f32(false, a, false, b, (short)0, c,
                                               false, false);
}

// GEMM: 16-row tile (rows rowBase..rowBase+15 of LDS act, stride lda)
// times W (K x ldw row-major, global), columns [n0, n0+16). K multiple of 4.
static __device__ __forceinline__ v8f gemm_lds(const float* act, int lda, int rowBase,
                                               const float* __restrict__ W, int ldw,
                                               int n0, int K, int lane) {
  const int half = lane >> 4;
  const int r    = lane & 15;
  const float* arow = act + (rowBase + r) * lda;
  v8f acc = {};
  for (int k0 = 0; k0 < K; k0 += 4) {
    v2f a, b;
    a.x = arow[k0 + 2 * half];
    a.y = arow[k0 + 2 * half + 1];
    b.x = W[(k0 + 2 * half) * ldw + n0 + r];
    b.y = W[(k0 + 2 * half + 1) * ldw + n0 + r];
    acc = wmma4(acc, a, b);
  }
  return acc;
}

// K=3 variant (act padded to stride 4 with zeros in col 3; W has 3 rows)
static __device__ __forceinline__ v8f gemm_k3(const float* act4, int rowBase,
                                              const float* __restrict__ W, int ldw,
                                              int n0, int lane) {
  const int half = lane >> 4;
  const int r    = lane & 15;
  const float* arow = act4 + (rowBase + r) * 4;
  const int k0 = 2 * half, k1 = 2 * half + 1;
  v2f a, b;
  a.x = arow[k0];
  a.y = arow[k1];                    // arow[3] == 0
  b.x = W[k0 * ldw + n0 + r];
  b.y = (k1 < 3) ? W[k1 * ldw + n0 + r] : 0.0f;
  v8f acc = {};
  return wmma4(acc, a, b);
}

static __device__ __forceinline__ void store_tile(float* dst, int ldd, int rowBase,
                                                  int n0, v8f acc,
                                                  const float* __restrict__ bias,
                                                  bool relu, int lane) {
  const int half = lane >> 4;
  const int r    = lane & 15;
  const float bv = bias[n0 + r];
#pragma unroll
  for (int v = 0; v < 8; ++v) {
    float val = acc[v] + bv;
    if (relu) val = fmaxf(val, 0.0f);
    dst[(rowBase + v + 8 * half) * ldd + n0 + r] = val;
  }
}

// column-wise max of a 16x16 tile (over the 16 rows), optional bias/relu.
// result valid in lanes 0..15 (column n0 + lane).
static __device__ __forceinline__ float tile_colmax(v8f acc, float bias, bool relu,
                                                    int lane) {
  float m = -3.0e38f;
#pragma unroll
  for (int v = 0; v < 8; ++v) {
    float val = acc[v] + bias;
    if (relu) val = fmaxf(val, 0.0f);
    m = fmaxf(m, val);
  }
  float o = __shfl_xor(m, 16, 32);
  return fmaxf(m, o);
}

// ---------------- Kernel A: farthest point sampling ----------------
// one block per batch; 1024 threads hold 16 points each in registers.
__global__ __launch_bounds__(1024) void fps_kernel(const float* __restrict__ pc,
                                                   float* __restrict__ newxyz) {
  const int b   = blockIdx.x;
  const int tid = threadIdx.x;
  const float* base = pc + (size_t)b * NN * 3;

  float px[16], py[16], pz[16], mind[16];
#pragma unroll
  for (int k = 0; k < 16; ++k) {
    const int i = k * 1024 + tid;
    px[k] = base[i * 3 + 0];
    py[k] = base[i * 3 + 1];
    pz[k] = base[i * 3 + 2];
    mind[k] = 1e10f;
  }

  __shared__ float redv[32];
  __shared__ int   redi[32];
  __shared__ float bx, by, bz;
  if (tid == 0) { bx = base[0]; by = base[1]; bz = base[2]; }
  __syncthreads();

  for (int t = 0; t < NPT; ++t) {
    if (tid == 0) {
      float* o = newxyz + ((size_t)b * NPT + t) * 3;
      o[0] = bx; o[1] = by; o[2] = bz;
    }
    const float lx = bx, ly = by, lz = bz;

    float bestv = -1.0f;
    int   besti = 0;
#pragma unroll
    for (int k = 0; k < 16; ++k) {
      const float dx = px[k] - lx, dy = py[k] - ly, dz = pz[k] - lz;
      const float d  = dx * dx + dy * dy + dz * dz;
      const float mv = fminf(mind[k], d);
      mind[k] = mv;
      const int i = k * 1024 + tid;
      if (mv > bestv) { bestv = mv; besti = i; }   // k ascending => first index kept
    }
    // intra-wave reduce (max value, smallest index on ties)
    for (int off = 16; off > 0; off >>= 1) {
      const float ov = __shfl_down(bestv, off, 32);
      const int   oi = __shfl_down(besti, off, 32);
      if (ov > bestv || (ov == bestv && oi < besti)) { bestv = ov; besti = oi; }
    }
    if ((tid & 31) == 0) { redv[tid >> 5] = bestv; redi[tid >> 5] = besti; }
    __syncthreads();
    if (tid < 32) {
      float v = redv[tid];
      int   i = redi[tid];
      for (int off = 16; off > 0; off >>= 1) {
        const float ov = __shfl_down(v, off, 32);
        const int   oi = __shfl_down(i, off, 32);
        if (ov > v || (ov == v && oi < i)) { v = ov; i = oi; }
      }
      if (tid == 0) {
        const float* p = base + (size_t)i * 3;
        bx = p[0]; by = p[1]; bz = p[2];
      }
    }
    __syncthreads();
  }
}

// ---------------- Kernel B: ball query + grouping ----------------
// 8 warps per block, one warp per keypoint.
__global__ __launch_bounds__(256) void bq_kernel(const float* __restrict__ pc,
                                                 const float* __restrict__ newxyz,
                                                 float* __restrict__ g) {
  const int warp = threadIdx.x >> 5;
  const int lane = threadIdx.x & 31;
  const int kp   = blockIdx.x * 8 + warp;   // kp = b*512 + m
  const int b    = kp >> 9;
  const float* base = pc + (size_t)b * NN * 3;

  __shared__ int idxbuf[8][NSAMP];

  const float cx = newxyz[(size_t)kp * 3 + 0];
  const float cy = newxyz[(size_t)kp * 3 + 1];
  const float cz = newxyz[(size_t)kp * 3 + 2];
  const float cc = (cx * cx + cy * cy) + cz * cz;

  int cnt = 0;
  for (int c0 = 0; c0 < NN; c0 += 32) {
    const int j = c0 + lane;
    const float x = base[j * 3 + 0];
    const float y = base[j * 3 + 1];
    const float z = base[j * 3 + 2];
    const float xx  = (x * x + y * y) + z * z;
    const float dot = (cx * x + cy * y) + cz * z;
    const float d2  = (cc + xx) - 2.0f * dot;      // same formula as reference
    const bool in = d2 < 4.0f;                     // RADIUS^2
    const unsigned bal = __builtin_amdgcn_ballot_w32(in);
    const int slot = cnt + __popc(bal & ((1u << lane) - 1u));
    if (in && slot < NSAMP) idxbuf[warp][slot] = j;
    cnt += __popc(bal);
    if (cnt >= NSAMP) break;
  }
  __syncthreads();
  if (cnt == 0 && lane == 0) idxbuf[warp][0] = 0;
  __syncthreads();
  const int first = idxbuf[warp][0];
  for (int s = lane; s < NSAMP; s += 32)
    if (s >= cnt) idxbuf[warp][s] = first;
  __syncthreads();
  for (int s = lane; s < NSAMP; s += 32) {
    const int j = idxbuf[warp][s];
    float* go = g + ((size_t)kp * NSAMP + s) * 3;
    go[0] = (base[j * 3 + 0] - cx) * RAD_INV;
    go[1] = (base[j * 3 + 1] - cy) * RAD_INV;
    go[2] = (base[j * 3 + 2] - cz) * RAD_INV;
  }
}

// ---------------- Kernel C: detector MLP (conv0..conv4, attention, orientation) --
// one block (4 waves) per keypoint; 16 rows per wave.
__global__ __launch_bounds__(128) void mlp1_kernel(
    const float* __restrict__ g,
    const float* __restrict__ dw0, const float* __restrict__ db0,
    const float* __restrict__ dw1, const float* __restrict__ db1,
    const float* __restrict__ dw2, const float* __restrict__ db2,
    const float* __restrict__ dw3, const float* __restrict__ db3,
    const float* __restrict__ dw4, const float* __restrict__ db4,
    const float* __restrict__ aw,  const float* __restrict__ ab,
    const float* __restrict__ ow,  const float* __restrict__ ob,
    float* __restrict__ att_out, float* __restrict__ cs) {
  const int kp   = blockIdx.x;
  const int tid  = threadIdx.x;
  const int lane = tid & 31;
  const int w    = tid >> 5;
  const int rowBase = w * 16;

  __shared__ float gbuf[64 * 4];
  __shared__ float bufA[64 * 64];
  __shared__ float bufB[64 * 128];
  __shared__ float xpart[4][256];
  __shared__ float xmax[256];
  __shared__ float x3[128];
  __shared__ float x4[64];

  const float* gp = g + (size_t)kp * NSAMP * 3;
  for (int i = tid; i < 64; i += 128) {
    gbuf[i * 4 + 0] = gp[i * 3 + 0];
    gbuf[i * 4 + 1] = gp[i * 3 + 1];
    gbuf[i * 4 + 2] = gp[i * 3 + 2];
    gbuf[i * 4 + 3] = 0.0f;
  }
  __syncthreads();

  // conv0: 3 -> 64, relu
  for (int nt = 0; nt < 4; ++nt) {
    v8f acc = gemm_k3(gbuf, rowBase, dw0, 64, nt * 16, lane);
    store_tile(bufA, 64, rowBase, nt * 16, acc, db0, true, lane);
  }
  __syncthreads();

  // conv1: 64 -> 128, relu
  for (int nt = 0; nt < 8; ++nt) {
    v8f acc = gemm_lds(bufA, 64, rowBase, dw1, 128, nt * 16, 64, lane);
    store_tile(bufB, 128, rowBase, nt * 16, acc, db1, true, lane);
  }
  __syncthreads();

  // conv2: 128 -> 256, relu, fused max over the 64 samples
  for (int nt = 0; nt < 16; ++nt) {
    v8f acc = gemm_lds(bufB, 128, rowBase, dw2, 256, nt * 16, 128, lane);
    const int r = lane & 15;
    const float cm = tile_colmax(acc, db2[nt * 16 + r], true, lane);
    if (lane < 16) xpart[w][nt * 16 + r] = cm;
  }
  __syncthreads();
  for (int c = tid; c < 256; c += 128)
    xmax[c] = fmaxf(fmaxf(xpart[0][c], xpart[1][c]),
                    fmaxf(xpart[2][c], xpart[3][c]));
  __syncthreads();

  // conv3: 256 -> 128, relu (single row)
  {
    float a = db3[tid];
    for (int k = 0; k < 256; ++k) a += xmax[k] * dw3[k * 128 + tid];
    x3[tid] = fmaxf(a, 0.0f);
  }
  __syncthreads();
  // conv4: 128 -> 64, relu
  if (tid < 64) {
    float a = db4[tid];
    for (int k = 0; k < 128; ++k) a += x3[k] * dw4[k * 64 + tid];
    x4[tid] = fmaxf(a, 0.0f);
  }
  __syncthreads();
  if (tid == 0) {
    float at = ab[0], oxv = ob[0], oyv = ob[1];
    for (int k = 0; k < 64; ++k) {
      const float v = x4[k];
      at  += v * aw[k];
      oxv += v * ow[k * 2 + 0];
      oyv += v * ow[k * 2 + 1];
    }
    // softplus = max(x,0) + log1p(exp(-|x|))
    att_out[kp] = fmaxf(at, 0.0f) + log1pf(expf(-fabsf(at)));
    const float nn = oxv * oxv + oyv * oyv;
    const float rs = rsqrtf(fmaxf(nn, EPSV));
    const float nx = oxv * rs, ny = oyv * rs;
    const float th = atan2f(ny, nx);
    cs[kp * 2 + 0] = cosf(th);
    cs[kp * 2 + 1] = sinf(th);
  }
}

// ---------------- Kernel D: descriptor MLP (rotate, ew0..ew3, l2norm) ----------
__global__ __launch_bounds__(128) void mlp2_kernel(
    const float* __restrict__ g, const float* __restrict__ cs,
    const float* __restrict__ ew0, const float* __restrict__ eb0,
    const float* __restrict__ ew1, const float* __restrict__ eb1,
    const float* __restrict__ ew2, const float* __restrict__ eb2,
    const float* __restrict__ ew3, const float* __restrict__ eb3,
    float* __restrict__ feats) {
  const int kp   = blockIdx.x;
  const int tid  = threadIdx.x;
  const int lane = tid & 31;
  const int w    = tid >> 5;
  const int rowBase = w * 16;

  __shared__ float g2buf[64 * 4];
  __shared__ float y0[64 * 32];
  __shared__ float y1[64 * 64];
  __shared__ float mpart[4][64];
  __shared__ float mbuf[64];
  __shared__ float basebuf[128];
  __shared__ float tpart[4][128];
  __shared__ float ymaxb[128];
  __shared__ float y3[64];
  __shared__ float srs;

  const float cv = cs[kp * 2 + 0];
  const float sv = cs[kp * 2 + 1];
  const float* gp = g + (size_t)kp * NSAMP * 3;
  for (int i = tid; i < 64; i += 128) {
    const float gx = gp[i * 3 + 0], gy = gp[i * 3 + 1], gz = gp[i * 3 + 2];
    g2buf[i * 4 + 0] = cv * gx + sv * gy;
    g2buf[i * 4 + 1] = -sv * gx + cv * gy;
    g2buf[i * 4 + 2] = gz;
    g2buf[i * 4 + 3] = 0.0f;
  }
  __syncthreads();

  // ew0: 3 -> 32, relu
  for (int nt = 0; nt < 2; ++nt) {
    v8f acc = gemm_k3(g2buf, rowBase, ew0, 32, nt * 16, lane);
    store_tile(y0, 32, rowBase, nt * 16, acc, eb0, true, lane);
  }
  __syncthreads();

  // ew1: 32 -> 64, relu, plus fused per-column max -> m
  for (int nt = 0; nt < 4; ++nt) {
    v8f acc = gemm_lds(y0, 32, rowBase, ew1, 64, nt * 16, 32, lane);
    store_tile(y1, 64, rowBase, nt * 16, acc, eb1, true, lane);
    const int r = lane & 15;
    const float cm = tile_colmax(acc, eb1[nt * 16 + r], true, lane);
    if (lane < 16) mpart[w][nt * 16 + r] = cm;
  }
  __syncthreads();
  if (tid < 64)
    mbuf[tid] = fmaxf(fmaxf(mpart[0][tid], mpart[1][tid]),
                      fmaxf(mpart[2][tid], mpart[3][tid]));
  __syncthreads();

  // base[n] = eb2[n] + m . ew2[0:64, n]  (broadcast half of the concat)
  {
    float a = eb2[tid];
    for (int k = 0; k < 64; ++k) a += mbuf[k] * ew2[k * 128 + tid];
    basebuf[tid] = a;
  }
  // t = y1 @ ew2[64:128, :], fused max over rows
  const float* W2b = ew2 + 64 * 128;
  for (int nt = 0; nt < 8; ++nt) {
    v8f acc = gemm_lds(y1, 64, rowBase, W2b, 128, nt * 16, 64, lane);
    const float cm = tile_colmax(acc, 0.0f, false, lane);
    if (lane < 16) tpart[w][nt * 16 + (lane & 15)] = cm;
  }
  __syncthreads();
  ymaxb[tid] = basebuf[tid] + fmaxf(fmaxf(tpart[0][tid], tpart[1][tid]),
                                    fmaxf(tpart[2][tid], tpart[3][tid]));
  __syncthreads();

  // ew3: 128 -> 64 (no activation)
  if (tid < 64) {
    float a = eb3[tid];
    for (int k = 0; k < 128; ++k) a += ymaxb[k] * ew3[k * 64 + tid];
    y3[tid] = a;
  }
  __syncthreads();
  if (tid == 0) {
    float ss = 0.0f;
    for (int k = 0; k < 64; ++k) ss += y3[k] * y3[k];
    srs = rsqrtf(fmaxf(ss, EPSV));
  }
  __syncthreads();
  if (tid < 64) feats[(size_t)kp * 64 + tid] = y3[tid] * srs;
}

// ---------------- launcher ----------------
extern "C" void kernel_launch(void* const* d_in, const int* in_sizes, int n_in,
                              void* d_out, int out_size, void* d_ws, size_t ws_size,
                              hipStream_t stream) {
  const float* pc  = (const float*)d_in[0];
  const float* dw0 = (const float*)d_in[1];
  const float* db0 = (const float*)d_in[2];
  const float* dw1 = (const float*)d_in[3];
  const float* db1 = (const float*)d_in[4];
  const float* dw2 = (const float*)d_in[5];
  const float* db2 = (const float*)d_in[6];
  const float* dw3 = (const float*)d_in[7];
  const float* db3 = (const float*)d_in[8];
  const float* dw4 = (const float*)d_in[9];
  const float* db4 = (const float*)d_in[10];
  const float* aw  = (const float*)d_in[11];
  const float* ab  = (const float*)d_in[12];
  const float* ow  = (const float*)d_in[13];
  const float* ob  = (const float*)d_in[14];
  const float* ew0 = (const float*)d_in[15];
  const float* eb0 = (const float*)d_in[16];
  const float* ew1 = (const float*)d_in[17];
  const float* eb1 = (const float*)d_in[18];
  const float* ew2 = (const float*)d_in[19];
  const float* eb2 = (const float*)d_in[20];
  const float* ew3 = (const float*)d_in[21];
  const float* eb3 = (const float*)d_in[22];

  float* out    = (float*)d_out;
  float* newxyz = out + OFF_NEWXYZ;
  float* feats  = out + OFF_FEATS;
  float* att    = out + OFF_ATT;

  float* gws = (float*)d_ws;                          // 8*512*64*3 floats
  float* csw = gws + (size_t)BB * NPT * NSAMP * 3;    // 8*512*2 floats

  fps_kernel<<<dim3(BB), dim3(1024), 0, stream>>>(pc, newxyz);
  bq_kernel<<<dim3(BB * NPT / 8), dim3(256), 0, stream>>>(pc, newxyz, gws);
  mlp1_kernel<<<dim3(BB * NPT), dim3(128), 0, stream>>>(
      gws, dw0, db0, dw1, db1, dw2, db2, dw3, db3, dw4, db4, aw, ab, ow, ob,
      att, csw);
  mlp2_kernel<<<dim3(BB * NPT), dim3(128), 0, stream>>>(
      gws, csw, ew0, eb0, ew1, eb1, ew2, eb2, ew3, eb3, feats);
}